// VQN_73486890434727
// MI455X (gfx1250) — compile-verified
//
#include <hip/hip_runtime.h>

#define BATCH   131072
#define IN_DIM  1153
#define CODE    16
#define OUT_DIM 768

#define K_PAD   1156   // IN_DIM rounded up to multiple of 4 (tail zero-padded)
#define O_PAD   772    // OUT_DIM + 4 : kills LDS bank conflicts on transpose

#define THREADS 256
#define WAVES   8
#define BLOCKS  512
#define TILES_PER_WAVE 2
#define TOTAL_WAVES (BLOCKS * WAVES)      // 4096 waves * 2 tiles = 8192 tiles = BATCH/16

typedef __attribute__((ext_vector_type(2))) float v2f;
typedef __attribute__((ext_vector_type(4))) float v4f;
typedef __attribute__((ext_vector_type(8))) float v8f;

__global__ __launch_bounds__(THREADS)
void vq_enc_dec_kernel(const float* __restrict__ x,
                       const float* __restrict__ W_enc,
                       const float* __restrict__ W_dec,
                       float* __restrict__ y)
{
    extern __shared__ float lds[];
    float* sEnc  = lds;                  // [CODE][K_PAD], rows zero-padded past IN_DIM
    float* sDecT = lds + CODE * K_PAD;   // [CODE][O_PAD], W_dec transposed

    // ---- stage W_enc into LDS (zero pad K to K_PAD) ----
    for (int i = threadIdx.x; i < CODE * K_PAD; i += THREADS) {
        int r = i / K_PAD;
        int k = i - r * K_PAD;
        sEnc[i] = (k < IN_DIM) ? W_enc[r * IN_DIM + k] : 0.0f;
    }
    // ---- stage W_dec transposed: sDecT[c][o] = W_dec[o][c] ----
    for (int i = threadIdx.x; i < OUT_DIM * CODE; i += THREADS) {
        int o = i >> 4;
        int c = i & (CODE - 1);
        sDecT[c * O_PAD + o] = W_dec[i];
    }
    __syncthreads();

    const int lane = threadIdx.x & 31;
    const int wave = threadIdx.x >> 5;
    const int rsub = lane & 15;   // A: row-in-tile / B: code index n
    const int kg   = lane >> 4;   // K-pair group (holds K = 2*kg, 2*kg+1 of each chunk)

    // B operand source: W_enc row 'rsub' (B[k][n] = W_enc[n][k]), 8B-aligned (K_PAD%4==0)
    const float* sB = sEnc + rsub * K_PAD;

    const int wgid = blockIdx.x * WAVES + wave;

    for (int t = 0; t < TILES_PER_WAVE; ++t) {
        const int tile = wgid + t * TOTAL_WAVES;            // 0..8191
        const float* xr = x + (size_t)(tile * 16 + rsub) * IN_DIM;

        v8f acc = {0.f, 0.f, 0.f, 0.f, 0.f, 0.f, 0.f, 0.f};

        // ---- main K loop: 288 chunks of 4 (K = 0..1151) ----
        for (int k0 = 0; k0 < 1152; k0 += 32) {
            __builtin_prefetch(xr + k0 + 96, 0, 0);         // global_prefetch_b8, ~384B ahead
            #pragma unroll
            for (int kk = 0; kk < 32; kk += 4) {
                const int k = k0 + kk + 2 * kg;
                v2f a; a.x = xr[k]; a.y = xr[k + 1];        // 2x b32 (row stride odd)
                v2f b = *(const v2f*)(sB + k);              // ds_load_b64, conflict-free
                acc = __builtin_amdgcn_wmma_f32_16x16x4_f32(
                        false, a, false, b, (short)0, acc, false, false);
            }
        }
        // ---- tail chunk: only K=1152 valid; B side zero-padded in LDS ----
        {
            v2f a = {0.f, 0.f};
            if (kg == 0) a.x = xr[1152];
            v2f b = *(const v2f*)(sB + 1152 + 2 * kg);
            acc = __builtin_amdgcn_wmma_f32_16x16x4_f32(
                    false, a, false, b, (short)0, acc, false, false);
        }

        // ---- argmax over code dim (16 lanes of a half-group hold row m = v + 8*kg) ----
        int codev[8];
        #pragma unroll
        for (int v = 0; v < 8; ++v) {
            float val = acc[v];
            int   idx = rsub;
            #pragma unroll
            for (int m = 1; m <= 8; m <<= 1) {
                float oval = __shfl_xor(val, m, 32);
                int   oidx = __shfl_xor(idx, m, 32);
                if (oval > val || (oval == val && oidx < idx)) { val = oval; idx = oidx; }
            }
            codev[v] = idx;   // lowest index on ties == jnp.argmax
        }

        // ---- decode = row gather: y[row] = W_decT[code], wave-cooperative NT stores ----
        #pragma unroll
        for (int m = 0; m < 16; ++m) {
            const int code = __shfl(codev[m & 7], (m >> 3) << 4, 32); // lane 0 or 16
            const v4f* src = (const v4f*)(sDecT + code * O_PAD);
            v4f* dst = (v4f*)(y + (size_t)(tile * 16 + m) * OUT_DIM);
            #pragma unroll
            for (int j = 0; j < 6; ++j) {                   // 192 float4 / row, 32 lanes
                v4f val = src[j * 32 + lane];
                __builtin_nontemporal_store(val, dst + j * 32 + lane);
            }
        }
    }
}

extern "C" void kernel_launch(void* const* d_in, const int* in_sizes, int n_in,
                              void* d_out, int out_size, void* d_ws, size_t ws_size,
                              hipStream_t stream) {
    const float* x     = (const float*)d_in[0];
    const float* W_enc = (const float*)d_in[1];
    const float* W_dec = (const float*)d_in[2];
    float* y = (float*)d_out;

    const size_t shmem = (size_t)(CODE * K_PAD + CODE * O_PAD) * sizeof(float); // 123392 B
    (void)hipFuncSetAttribute((const void*)vq_enc_dec_kernel,
                              hipFuncAttributeMaxDynamicSharedMemorySize, (int)shmem);
    vq_enc_dec_kernel<<<BLOCKS, THREADS, shmem, stream>>>(x, W_enc, W_dec, y);
}